// GINe_60997125538471
// MI455X (gfx1250) — compile-verified
//
#include <hip/hip_runtime.h>
#include <math.h>

// ---------------- problem constants (match reference) ----------------
#define NN   100000   // nodes
#define EE   1000000  // message edges
#define EPN  100000   // pos/neg link-pred edges
#define BN_EPS 1e-5f

// ---------------- WMMA / vector types ----------------
typedef float v2f __attribute__((ext_vector_type(2)));
typedef float v8f __attribute__((ext_vector_type(8)));
typedef int   v4i __attribute__((ext_vector_type(4)));

#define AS1 __attribute__((address_space(1)))
#define AS3 __attribute__((address_space(3)))

#if defined(__has_builtin)
#if __has_builtin(__builtin_amdgcn_global_load_async_to_lds_b128) && \
    __has_builtin(__builtin_amdgcn_s_wait_asynccnt)
#define HAVE_ASYNC_LDS 1
#endif
#endif
#ifndef HAVE_ASYNC_LDS
#define HAVE_ASYNC_LDS 0
#endif

// ---------------- generic 64-col linear layer via V_WMMA_F32_16X16X4_F32 ----
// C[M,64] = op(A)[M,K] @ W[K,64] + bias, optional ReLU epilogue.
// MODE 0: A = A1                              (async global->LDS staging)
// MODE 1: A = A1 + A2                         (fuses z = h + agg)
// MODE 2: A row r = [relu(h[src[r]]) | relu(h[dst[r]]) | Eenc[r]], K = 192
#define TILE_ROWS 128
#define KCHUNK    32
#define LDA       36    // KCHUNK + 4: 144B row stride (16B aligned), bank-spread
#define PSTR      162   // W pair stride in floats: 64 cols * 2 + 34 pad

template <int MODE, bool RELU>
__global__ __launch_bounds__(256) void lin64_wmma(
    const float* __restrict__ A1, const float* __restrict__ A2,
    const int* __restrict__ srcIdx, const int* __restrict__ dstIdx,
    const float* __restrict__ Eenc,
    const float* __restrict__ W, const float* __restrict__ bias,
    float* __restrict__ C, int M, int K)
{
    __shared__ float sA[TILE_ROWS * LDA];        // 18432 B
    __shared__ float sW[(KCHUNK / 2) * PSTR];    // 10368 B, K-pair interleaved
    __shared__ float sBias[64];

    const int t    = threadIdx.x;
    const int wave = t >> 5;
    const int lane = t & 31;
    const int half = lane >> 4;
    const int m16  = lane & 15;
    const int blockRow = blockIdx.x * TILE_ROWS;

    if (t < 64) sBias[t] = bias[t];

    v8f acc[4] = {};   // 16x64 output tile per wave (4 n-tiles of 16)

    for (int kk = 0; kk < K; kk += KCHUNK) {
        __syncthreads();
        // ---- stage W chunk [32 x 64] into LDS, K-pair interleaved:
        //      sW[(p*PSTR) + col*2 + parity] = W[kk + 2p + parity][col]
        {
            int r   = t >> 3;            // 0..31
            int q   = (t & 7) * 8;       // col 0,8,..,56
            const float4* wp = (const float4*)(W + (size_t)(kk + r) * 64 + q);
            float4 w0 = wp[0], w1 = wp[1];
            float* d = &sW[(r >> 1) * PSTR + q * 2 + (r & 1)];
            d[0]  = w0.x; d[2]  = w0.y; d[4]  = w0.z; d[6]  = w0.w;
            d[8]  = w1.x; d[10] = w1.y; d[12] = w1.z; d[14] = w1.w;
        }
        // ---- stage A chunk [128 x 32] into LDS ----
        #pragma unroll
        for (int it = 0; it < 4; ++it) {
            int slot = t + it * 256;        // 0..1023
            int r    = slot >> 3;           // block row 0..127
            int q    = (slot & 7) * 4;      // chunk col 0..28
            int grow = blockRow + r;
            if (MODE == 0) {
#if HAVE_ASYNC_LDS
                // async DMA global -> LDS (ASYNCcnt); rows >= M skipped: a stale
                // A row only feeds its own (unstored) output row.
                if (grow < M) {
                    __builtin_amdgcn_global_load_async_to_lds_b128(
                        (AS1 v4i*)(A1 + (size_t)grow * K + kk + q),
                        (AS3 v4i*)&sA[r * LDA + q], 0, 0);
                }
#else
                float4 v = make_float4(0.f, 0.f, 0.f, 0.f);
                if (grow < M) v = *(const float4*)(A1 + (size_t)grow * K + kk + q);
                float* d = &sA[r * LDA + q];
                d[0] = v.x; d[1] = v.y; d[2] = v.z; d[3] = v.w;
#endif
            } else {
                float4 v = make_float4(0.f, 0.f, 0.f, 0.f);
                if (grow < M) {
                    if (MODE == 1) {
                        float4 a = *(const float4*)(A1 + (size_t)grow * K + kk + q);
                        float4 b = *(const float4*)(A2 + (size_t)grow * K + kk + q);
                        v = make_float4(a.x + b.x, a.y + b.y, a.z + b.z, a.w + b.w);
                    } else { // MODE 2: gather head features
                        int col = kk + q;
                        if (col < 64) {
                            int s = srcIdx[grow];
                            float4 hv = *(const float4*)(A1 + (size_t)s * 64 + col);
                            v = make_float4(fmaxf(hv.x, 0.f), fmaxf(hv.y, 0.f),
                                            fmaxf(hv.z, 0.f), fmaxf(hv.w, 0.f));
                        } else if (col < 128) {
                            int d2 = dstIdx[grow];
                            float4 hv = *(const float4*)(A1 + (size_t)d2 * 64 + (col - 64));
                            v = make_float4(fmaxf(hv.x, 0.f), fmaxf(hv.y, 0.f),
                                            fmaxf(hv.z, 0.f), fmaxf(hv.w, 0.f));
                        } else {
                            v = *(const float4*)(Eenc + (size_t)grow * 64 + (col - 128));
                        }
                    }
                }
                float* d = &sA[r * LDA + q];
                d[0] = v.x; d[1] = v.y; d[2] = v.z; d[3] = v.w;
            }
        }
#if HAVE_ASYNC_LDS
        if (MODE == 0) __builtin_amdgcn_s_wait_asynccnt(0);
#endif
        __syncthreads();

        // ---- 8 WMMA k-steps over this chunk ----
        const int arow = (wave * 16 + m16) * LDA;
        #pragma unroll
        for (int ks = 0; ks < KCHUNK; ks += 4) {
            // A frag: lanes 0-15 hold {K=ks, ks+1}, lanes 16-31 hold {ks+2, ks+3}
            v2f a = *(const v2f*)&sA[arow + ks + 2 * half];
            // B frag: one aligned b64 per n-tile from the interleaved pair layout
            const int pbase = ((ks >> 1) + half) * PSTR + m16 * 2;
            #pragma unroll
            for (int nt = 0; nt < 4; ++nt) {
                v2f b = *(const v2f*)&sW[pbase + nt * 32];
                acc[nt] = __builtin_amdgcn_wmma_f32_16x16x4_f32(
                    false, a, false, b, (short)0, acc[nt], false, false);
            }
        }
    }

    // ---- epilogue: bias (+ReLU) + store. row = base + r + 8*half, col = nt*16+m16
    int rowBase = blockRow + wave * 16 + half * 8;
    #pragma unroll
    for (int r = 0; r < 8; ++r) {
        int row = rowBase + r;
        if (row < M) {
            #pragma unroll
            for (int nt = 0; nt < 4; ++nt) {
                int col = nt * 16 + m16;
                float v = acc[nt][r] + sBias[col];
                if (RELU) v = fmaxf(v, 0.f);
                C[(size_t)row * 64 + col] = v;
            }
        }
    }
}

// ---------------- message + scatter: agg[dst] += relu(h[src] + ea) ----------
__global__ __launch_bounds__(256) void k_message(
    const float* __restrict__ h, const float* __restrict__ ea,
    const int* __restrict__ src, const int* __restrict__ dst,
    float* __restrict__ agg, int E)
{
    int idx = blockIdx.x * 256 + threadIdx.x;     // E*16 threads, 4 cols each
    if (idx >= E * 16) return;
    int e = idx >> 4;
    int c = (idx & 15) << 2;
    int s = src[e], d = dst[e];
    float4 hv = *(const float4*)(h + (size_t)s * 64 + c);
    float4 ev = *(const float4*)(ea + (size_t)e * 64 + c);
    float* ap = agg + (size_t)d * 64 + c;
    atomicAdd(ap + 0, fmaxf(hv.x + ev.x, 0.f));
    atomicAdd(ap + 1, fmaxf(hv.y + ev.y, 0.f));
    atomicAdd(ap + 2, fmaxf(hv.z + ev.z, 0.f));
    atomicAdd(ap + 3, fmaxf(hv.w + ev.w, 0.f));
}

// ---------------- per-column sum / sum-of-squares for BatchNorm ------------
__global__ __launch_bounds__(256) void k_colstats(
    const float* __restrict__ z, float* __restrict__ stats, int M)
{
    __shared__ float ssum[4][64];
    __shared__ float ssq[4][64];
    int t = threadIdx.x, g = t >> 6, c = t & 63;
    float s = 0.f, q = 0.f;
    for (int row = blockIdx.x * 4 + g; row < M; row += gridDim.x * 4) {
        float v = z[(size_t)row * 64 + c];
        s += v; q += v * v;
    }
    ssum[g][c] = s; ssq[g][c] = q;
    __syncthreads();
    if (t < 64) {
        float S = 0.f, Q = 0.f;
        #pragma unroll
        for (int gg = 0; gg < 4; ++gg) { S += ssum[gg][t]; Q += ssq[gg][t]; }
        atomicAdd(&stats[t], S);
        atomicAdd(&stats[64 + t], Q);
    }
}

// ---------------- BN (batch stats) + ReLU + residual average ---------------
__global__ __launch_bounds__(256) void k_bn_res(
    float* __restrict__ h, const float* __restrict__ z,
    const float* __restrict__ stats, const float* __restrict__ gamma,
    const float* __restrict__ beta, int M)
{
    int idx = blockIdx.x * 256 + threadIdx.x;
    if (idx >= M * 64) return;
    int c = idx & 63;
    float invM = 1.f / (float)M;
    float mean = stats[c] * invM;
    float var  = stats[64 + c] * invM - mean * mean;
    float zn = (z[idx] - mean) * rsqrtf(var + BN_EPS) * gamma[c] + beta[c];
    h[idx] = (h[idx] + fmaxf(zn, 0.f)) * 0.5f;
}

// ---------------- head tail: relu(f1@W2+b2) @ W3 + b3 -> sigmoid -----------
__global__ __launch_bounds__(256) void k_head_tail(
    const float* __restrict__ f1, const float* __restrict__ w2,
    const float* __restrict__ b2, const float* __restrict__ w3,
    const float* __restrict__ b3, float* __restrict__ out, int M)
{
    __shared__ float sw2[64 * 25];
    __shared__ float sb2[25];
    __shared__ float sw3[25];
    __shared__ float sb3;
    int t = threadIdx.x;
    for (int i = t; i < 64 * 25; i += 256) sw2[i] = w2[i];
    if (t < 25) { sb2[t] = b2[t]; sw3[t] = w3[t]; }
    if (t == 0) sb3 = b3[0];
    __syncthreads();
    int row = blockIdx.x * 256 + t;
    if (row >= M) return;
    float acc[25];
    #pragma unroll
    for (int j = 0; j < 25; ++j) acc[j] = sb2[j];
    const float* fr = f1 + (size_t)row * 64;
    for (int k = 0; k < 64; ++k) {
        float v = fr[k];
        #pragma unroll
        for (int j = 0; j < 25; ++j) acc[j] += v * sw2[k * 25 + j];
    }
    float o = sb3;
    #pragma unroll
    for (int j = 0; j < 25; ++j) o += fmaxf(acc[j], 0.f) * sw3[j];
    out[row] = 1.f / (1.f + expf(-o));
}

// ---------------- host-side orchestration ----------------------------------
extern "C" void kernel_launch(void* const* d_in, const int* in_sizes, int n_in,
                              void* d_out, int out_size, void* d_ws, size_t ws_size,
                              hipStream_t stream)
{
    const float* x              = (const float*)d_in[0];
    const int*   edge_index     = (const int*)  d_in[1];   // [2,E]: row0=src,row1=dst
    const float* edge_attr      = (const float*)d_in[2];
    const int*   pos_edge_index = (const int*)  d_in[3];
    const float* pos_edge_attr  = (const float*)d_in[4];
    const int*   neg_edge_index = (const int*)  d_in[5];
    const float* neg_edge_attr  = (const float*)d_in[6];
    const float* node_w = (const float*)d_in[7];
    const float* node_b = (const float*)d_in[8];
    const float* edge_w = (const float*)d_in[9];
    const float* edge_b = (const float*)d_in[10];
    const float* conv_w1 = (const float*)d_in[11];   // [L,64,64]
    const float* conv_b1 = (const float*)d_in[12];
    const float* conv_w2 = (const float*)d_in[13];
    const float* conv_b2 = (const float*)d_in[14];
    const float* bn_gamma = (const float*)d_in[15];
    const float* bn_beta  = (const float*)d_in[16];
    const float* mlp1_w = (const float*)d_in[17];    // [192,64]
    const float* mlp1_b = (const float*)d_in[18];
    const float* mlp2_w = (const float*)d_in[19];    // [64,25]
    const float* mlp2_b = (const float*)d_in[20];
    const float* mlp3_w = (const float*)d_in[21];    // [25,1]
    const float* mlp3_b = (const float*)d_in[22];

    // outputs: pos[EP], neg[EP], h[N*64]
    float* outPos = (float*)d_out;
    float* outNeg = outPos + EPN;
    float* h      = outNeg + EPN;

    // workspace layout (floats)
    float* ws   = (float*)d_ws;
    float* ea   = ws;                                // E*64   = 64,000,000
    float* pe   = ea + (size_t)EE * 64;              // EP*64
    float* ne   = pe + (size_t)EPN * 64;             // EP*64
    float* agg  = ne + (size_t)EPN * 64;             // N*64  (also z2)
    float* z1   = agg + (size_t)NN * 64;             // N*64  (also f1)
    float* bns  = z1 + (size_t)NN * 64;              // 128

    auto rb = [](int M) { return (M + TILE_ROWS - 1) / TILE_ROWS; };

    // encoders
    lin64_wmma<0, false><<<rb(NN), 256, 0, stream>>>(
        x, nullptr, nullptr, nullptr, nullptr, node_w, node_b, h, NN, 128);
    lin64_wmma<0, false><<<rb(EE), 256, 0, stream>>>(
        edge_attr, nullptr, nullptr, nullptr, nullptr, edge_w, edge_b, ea, EE, 32);
    lin64_wmma<0, false><<<rb(EPN), 256, 0, stream>>>(
        pos_edge_attr, nullptr, nullptr, nullptr, nullptr, edge_w, edge_b, pe, EPN, 32);
    lin64_wmma<0, false><<<rb(EPN), 256, 0, stream>>>(
        neg_edge_attr, nullptr, nullptr, nullptr, nullptr, edge_w, edge_b, ne, EPN, 32);

    // GINEConv layers
    for (int l = 0; l < 2; ++l) {
        (void)hipMemsetAsync(agg, 0, (size_t)NN * 64 * sizeof(float), stream);
        k_message<<<(EE * 16 + 255) / 256, 256, 0, stream>>>(
            h, ea, edge_index, edge_index + EE, agg, EE);
        // z1 = relu((h + agg) @ W1 + b1)
        lin64_wmma<1, true><<<rb(NN), 256, 0, stream>>>(
            h, agg, nullptr, nullptr, nullptr,
            conv_w1 + (size_t)l * 64 * 64, conv_b1 + l * 64, z1, NN, 64);
        // z2 = z1 @ W2 + b2   (reuse agg buffer)
        lin64_wmma<0, false><<<rb(NN), 256, 0, stream>>>(
            z1, nullptr, nullptr, nullptr, nullptr,
            conv_w2 + (size_t)l * 64 * 64, conv_b2 + l * 64, agg, NN, 64);
        (void)hipMemsetAsync(bns, 0, 128 * sizeof(float), stream);
        k_colstats<<<512, 256, 0, stream>>>(agg, bns, NN);
        k_bn_res<<<(NN * 64 + 255) / 256, 256, 0, stream>>>(
            h, agg, bns, bn_gamma + l * 64, bn_beta + l * 64, NN);
    }

    // link-prediction heads
    lin64_wmma<2, true><<<rb(EPN), 256, 0, stream>>>(
        h, nullptr, pos_edge_index, pos_edge_index + EPN, pe,
        mlp1_w, mlp1_b, z1, EPN, 192);
    k_head_tail<<<(EPN + 255) / 256, 256, 0, stream>>>(
        z1, mlp2_w, mlp2_b, mlp3_w, mlp3_b, outPos, EPN);

    lin64_wmma<2, true><<<rb(EPN), 256, 0, stream>>>(
        h, nullptr, neg_edge_index, neg_edge_index + EPN, ne,
        mlp1_w, mlp1_b, z1, EPN, 192);
    k_head_tail<<<(EPN + 255) / 256, 256, 0, stream>>>(
        z1, mlp2_w, mlp2_b, mlp3_w, mlp3_b, outNeg, EPN);
}